// Transformer_45148696216027
// MI455X (gfx1250) — compile-verified
//
#include <hip/hip_runtime.h>
#include <math.h>

// ---------------------------------------------------------------------------
// Transformer forward (B=8, N=512, E=512, H=8, F=2048, 2 enc + 2 dec layers)
// for MI455X / gfx1250.
//
//  * Decoder's size-1-axis LayerNorm makes the cross-attn query an exact
//    constant row q_vec = b1*Wl + bl (independent of y / mW*), so the decoder
//    collapses to per-batch matvecs (exact algebra, ~0.1 GFLOPs).
//  * Encoder: ~240 GFLOPs of f32 GEMMs on V_WMMA_F32_16X16X4_F32 (native f32
//    matrix pipe, full reference precision).
//  * NN GEMMs use GLOBAL_LOAD_ASYNC_TO_LDS_B128 double-buffering when the
//    builtin is available (ASYNCcnt-tracked, one barrier per K-tile, global->
//    LDS overlapped with WMMA); otherwise fall back to the proven
//    VGPR-staged single-buffer pipeline.
// ---------------------------------------------------------------------------

#define B_      8
#define N_      512
#define E_      512
#define H_      8
#define F_      2048
#define HE_     4096            // H_*E_
#define ROWS_   4096            // B_*N_
#define EPS_    1e-5f

typedef __attribute__((ext_vector_type(2))) float v2f;
typedef __attribute__((ext_vector_type(8))) float v8f;

#if defined(__has_builtin)
#if __has_builtin(__builtin_amdgcn_global_load_async_to_lds_b128) && \
    __has_builtin(__builtin_amdgcn_s_wait_asynccnt)
#define HAVE_ASYNC_LDS 1
#endif
#endif
#ifndef HAVE_ASYNC_LDS
#define HAVE_ASYNC_LDS 0
#endif

#if HAVE_ASYNC_LDS
// builtin parameter types: 16-byte int vector pointers in AS(1) / AS(3)
typedef int v4i_b128 __attribute__((vector_size(16)));
typedef __attribute__((address_space(1))) v4i_b128* async_gptr;
typedef __attribute__((address_space(3))) v4i_b128* async_lptr;
#endif

// ---------------------------- block reductions -----------------------------
__device__ __forceinline__ float blk_reduce_sum(float v, float* sbuf) {
  int t = threadIdx.x;
  sbuf[t] = v; __syncthreads();
  #pragma unroll
  for (int s = 128; s >= 1; s >>= 1) {
    if (t < s) sbuf[t] += sbuf[t + s];
    __syncthreads();
  }
  float r = sbuf[0]; __syncthreads();
  return r;
}
__device__ __forceinline__ float blk_reduce_max(float v, float* sbuf) {
  int t = threadIdx.x;
  sbuf[t] = v; __syncthreads();
  #pragma unroll
  for (int s = 128; s >= 1; s >>= 1) {
    if (t < s) sbuf[t] = fmaxf(sbuf[t], sbuf[t + s]);
    __syncthreads();
  }
  float r = sbuf[0]; __syncthreads();
  return r;
}

// ------------------------ f32 WMMA GEMM (16x16x4) --------------------------
// C[M,N] = A[M,K] @ B[K,N]            (NT=0, B row-major [K,N])
// C[M,N] = A[M,K] @ B^T, B=[N,K]      (NT=1)
// Optional: += bias[col], += resid[row,col], relu. Batched via blockIdx.z.
// Requires: M%128==0, N%64==0, K%32==0, full blocks (EXEC all ones for WMMA).
template <int NT>
__global__ __launch_bounds__(256) void gemm_f32_wmma(
    const float* __restrict__ A, long long sAz, int lda,
    const float* __restrict__ Bm, long long sBz, int ldb,
    float* __restrict__ C, long long sCz, int ldc,
    const float* __restrict__ bias,
    const float* __restrict__ resid, long long sRz, int ldr,
    int K, int relu)
{
  constexpr int TBM = 128, TBN = 64, TBK = 32;
  constexpr int LDA_S = TBK + 4;   // 36 floats: 16B-aligned rows, conflict-free
  constexpr int LDB_S = TBN + 4;   // 68 floats
  __shared__ float sA[TBM * LDA_S];
  __shared__ float sB[TBK * LDB_S];

  const int tid  = threadIdx.x;
  const int lane = tid & 31;
  const int wave = tid >> 5;
  const int wm = wave >> 1;            // 4 wave-rows
  const int wn = wave & 1;             // 2 wave-cols
  const int l15   = lane & 15;
  const int lhalf = lane >> 4;         // 0/1 -> K pair (A/B), +8 rows (C/D)
  const int kOff  = lhalf * 2;

  const int m0 = blockIdx.y * TBM;
  const int n0 = blockIdx.x * TBN;
  A  += (long long)blockIdx.z * sAz;
  Bm += (long long)blockIdx.z * sBz;
  C  += (long long)blockIdx.z * sCz;
  if (resid) resid += (long long)blockIdx.z * sRz;

  v8f acc[2][2] = {};

  for (int k0 = 0; k0 < K; k0 += TBK) {
    // --- load A tile: 128x32 = 1024 float4, 4 per thread ---
    #pragma unroll
    for (int i = 0; i < 4; ++i) {
      int s  = tid + i * 256;
      int r  = s >> 3;         // / (TBK/4)
      int c4 = s & 7;
      const float4 v = *(const float4*)(A + (long long)(m0 + r) * lda + k0 + c4 * 4);
      *(float4*)(sA + r * LDA_S + c4 * 4) = v;
    }
    // --- load B tile ---
    if (NT == 0) {             // B[K,N]: 32x64 = 512 float4, 2 per thread
      #pragma unroll
      for (int i = 0; i < 2; ++i) {
        int s  = tid + i * 256;
        int r  = s >> 4;       // / (TBN/4)
        int c4 = s & 15;
        const float4 v = *(const float4*)(Bm + (long long)(k0 + r) * ldb + n0 + c4 * 4);
        *(float4*)(sB + r * LDB_S + c4 * 4) = v;
      }
    } else {                   // B[N,K]: read float4 along K, scatter transposed
      #pragma unroll
      for (int i = 0; i < 2; ++i) {
        int s  = tid + i * 256;
        int n  = s >> 3;
        int c4 = s & 7;
        const float4 v = *(const float4*)(Bm + (long long)(n0 + n) * ldb + k0 + c4 * 4);
        sB[(c4 * 4 + 0) * LDB_S + n] = v.x;
        sB[(c4 * 4 + 1) * LDB_S + n] = v.y;
        sB[(c4 * 4 + 2) * LDB_S + n] = v.z;
        sB[(c4 * 4 + 3) * LDB_S + n] = v.w;
      }
    }
    __syncthreads();

    #pragma unroll
    for (int kk = 0; kk < TBK; kk += 4) {
      v2f af[2], bf[2];
      #pragma unroll
      for (int mi = 0; mi < 2; ++mi) {
        // ISA A 16x4 layout: lanes 0-15 -> K{kk,kk+1}, lanes 16-31 -> K{kk+2,kk+3}
        const float* p = sA + (wm * 32 + mi * 16 + l15) * LDA_S + kk + kOff;
        af[mi][0] = p[0];
        af[mi][1] = p[1];
      }
      #pragma unroll
      for (int ni = 0; ni < 2; ++ni) {
        // ISA B 4x16 layout (mirrored): lane half selects K pair, VGPR = K in pair
        const float* p = sB + (kk + kOff) * LDB_S + wn * 32 + ni * 16 + l15;
        bf[ni][0] = p[0];
        bf[ni][1] = p[LDB_S];
      }
      #pragma unroll
      for (int mi = 0; mi < 2; ++mi)
        #pragma unroll
        for (int ni = 0; ni < 2; ++ni)
          acc[mi][ni] = __builtin_amdgcn_wmma_f32_16x16x4_f32(
              false, af[mi], false, bf[ni], (short)0, acc[mi][ni], false, false);
    }
    __syncthreads();
  }

  // --- epilogue: C/D layout: VGPR r -> row r (lanes 0-15) / row r+8 (16-31) ---
  #pragma unroll
  for (int mi = 0; mi < 2; ++mi) {
    #pragma unroll
    for (int ni = 0; ni < 2; ++ni) {
      const int col = n0 + wn * 32 + ni * 16 + l15;
      #pragma unroll
      for (int r = 0; r < 8; ++r) {
        const int row = m0 + wm * 32 + mi * 16 + r + lhalf * 8;
        float v = acc[mi][ni][r];
        if (bias)  v += bias[col];
        if (resid) v += resid[(long long)row * ldr + col];
        if (relu)  v = fmaxf(v, 0.f);
        C[(long long)row * ldc + col] = v;
      }
    }
  }
}

#if HAVE_ASYNC_LDS
// NN GEMM with async global->LDS double buffering: prefetch of K-tile k+1
// (ASYNCcnt-tracked, no VGPR staging) overlaps WMMA on tile k; one barrier
// per tile (wait own asynccnt -> barrier => all deposits visible and all
// readers of the other buffer are done before it is overwritten).
__global__ __launch_bounds__(256) void gemm_nn_async(
    const float* __restrict__ A, long long sAz, int lda,
    const float* __restrict__ Bm, long long sBz, int ldb,
    float* __restrict__ C, long long sCz, int ldc,
    const float* __restrict__ bias,
    const float* __restrict__ resid, long long sRz, int ldr,
    int K, int relu)
{
  constexpr int TBM = 128, TBN = 64, TBK = 32;
  constexpr int LDA_S = TBK + 4;
  constexpr int LDB_S = TBN + 4;
  __shared__ float sA[2][TBM * LDA_S];
  __shared__ float sB[2][TBK * LDB_S];

  const int tid  = threadIdx.x;
  const int lane = tid & 31;
  const int wave = tid >> 5;
  const int wm = wave >> 1;
  const int wn = wave & 1;
  const int l15   = lane & 15;
  const int lhalf = lane >> 4;
  const int kOff  = lhalf * 2;

  const int m0 = blockIdx.y * TBM;
  const int n0 = blockIdx.x * TBN;
  A  += (long long)blockIdx.z * sAz;
  Bm += (long long)blockIdx.z * sBz;
  C  += (long long)blockIdx.z * sCz;
  if (resid) resid += (long long)blockIdx.z * sRz;

  auto issue_tile = [&](int bf, int k0) {
    #pragma unroll
    for (int i = 0; i < 4; ++i) {
      int s  = tid + i * 256;
      int r  = s >> 3;
      int c4 = s & 7;
      __builtin_amdgcn_global_load_async_to_lds_b128(
          (async_gptr)(A + (long long)(m0 + r) * lda + k0 + c4 * 4),
          (async_lptr)(&sA[bf][r * LDA_S + c4 * 4]), 0, 0);
    }
    #pragma unroll
    for (int i = 0; i < 2; ++i) {
      int s  = tid + i * 256;
      int r  = s >> 4;
      int c4 = s & 15;
      __builtin_amdgcn_global_load_async_to_lds_b128(
          (async_gptr)(Bm + (long long)(k0 + r) * ldb + n0 + c4 * 4),
          (async_lptr)(&sB[bf][r * LDB_S + c4 * 4]), 0, 0);
    }
  };

  v8f acc[2][2] = {};
  issue_tile(0, 0);
  int buf = 0;
  for (int k0 = 0; k0 < K; k0 += TBK) {
    __builtin_amdgcn_s_wait_asynccnt(0);   // own deposits for tile k0 done
    __syncthreads();                       // everyone's deposits visible;
                                           // everyone done reading buf^1
    if (k0 + TBK < K) issue_tile(buf ^ 1, k0 + TBK);

    #pragma unroll
    for (int kk = 0; kk < TBK; kk += 4) {
      v2f af[2], bf2[2];
      #pragma unroll
      for (int mi = 0; mi < 2; ++mi) {
        const float* p = &sA[buf][(wm * 32 + mi * 16 + l15) * LDA_S + kk + kOff];
        af[mi][0] = p[0];
        af[mi][1] = p[1];
      }
      #pragma unroll
      for (int ni = 0; ni < 2; ++ni) {
        const float* p = &sB[buf][(kk + kOff) * LDB_S + wn * 32 + ni * 16 + l15];
        bf2[ni][0] = p[0];
        bf2[ni][1] = p[LDB_S];
      }
      #pragma unroll
      for (int mi = 0; mi < 2; ++mi)
        #pragma unroll
        for (int ni = 0; ni < 2; ++ni)
          acc[mi][ni] = __builtin_amdgcn_wmma_f32_16x16x4_f32(
              false, af[mi], false, bf2[ni], (short)0, acc[mi][ni], false, false);
    }
    buf ^= 1;
  }

  #pragma unroll
  for (int mi = 0; mi < 2; ++mi) {
    #pragma unroll
    for (int ni = 0; ni < 2; ++ni) {
      const int col = n0 + wn * 32 + ni * 16 + l15;
      #pragma unroll
      for (int r = 0; r < 8; ++r) {
        const int row = m0 + wm * 32 + mi * 16 + r + lhalf * 8;
        float v = acc[mi][ni][r];
        if (bias)  v += bias[col];
        if (resid) v += resid[(long long)row * ldr + col];
        if (relu)  v = fmaxf(v, 0.f);
        C[(long long)row * ldc + col] = v;
      }
    }
  }
}
#endif  // HAVE_ASYNC_LDS

// ----------------------------- small kernels -------------------------------
__global__ __launch_bounds__(256) void posenc_add(const float* __restrict__ X,
                                                  float* __restrict__ O) {
  long long idx = (long long)blockIdx.x * 256 + threadIdx.x;   // B*N*E
  int e = (int)(idx & (E_ - 1));
  int n = (int)((idx >> 9) & (N_ - 1));
  float i2  = 2.0f * (float)e / (float)E_;
  float ang = (float)n * powf(1000.0f, -i2);
  float pe  = (e & 1) ? cosf(ang) : sinf(ang);
  O[idx] = X[idx] + pe;
}

// softmax over rows of length L (scale applied before max/exp)
__global__ __launch_bounds__(256) void softmax_rows(float* __restrict__ S, int L,
                                                    float scale) {
  __shared__ float sbuf[256];
  float* p = S + (long long)blockIdx.x * L;
  float mx = -3.4e38f;
  for (int j = threadIdx.x; j < L; j += 256) mx = fmaxf(mx, p[j] * scale);
  mx = blk_reduce_max(mx, sbuf);
  float sum = 0.f;
  for (int j = threadIdx.x; j < L; j += 256) {
    float e = expf(p[j] * scale - mx);
    p[j] = e;
    sum += e;
  }
  sum = blk_reduce_sum(sum, sbuf);
  float inv = 1.0f / sum;
  for (int j = threadIdx.x; j < L; j += 256) p[j] *= inv;
}

// LayerNorm over rows of length 512 (two elements per thread)
__global__ __launch_bounds__(256) void layernorm512(const float* __restrict__ X,
                                                    float* __restrict__ Y,
                                                    const float* __restrict__ g,
                                                    const float* __restrict__ b) {
  __shared__ float sbuf[256];
  const float* px = X + (long long)blockIdx.x * E_;
  float* py = Y + (long long)blockIdx.x * E_;
  int t = threadIdx.x;
  float x0 = px[t], x1 = px[t + 256];
  float mean = blk_reduce_sum(x0 + x1, sbuf) * (1.0f / (float)E_);
  float d0 = x0 - mean, d1 = x1 - mean;
  float var = blk_reduce_sum(d0 * d0 + d1 * d1, sbuf) * (1.0f / (float)E_);
  float inv = 1.0f / sqrtf(var + EPS_);
  py[t]       = d0 * inv * g[t]       + b[t];
  py[t + 256] = d1 * inv * g[t + 256] + b[t + 256];
}

// ---- decoder (algebraically collapsed: query row q_vec = b1*Wl + bl) ----
__global__ __launch_bounds__(256) void dec_qp(const float* __restrict__ Wq,
                                              const float* __restrict__ Wl,
                                              const float* __restrict__ bl,
                                              const float* __restrict__ b1,
                                              float* __restrict__ qp) {
  int idx = blockIdx.x * 256 + threadIdx.x;    // H*E
  int h = idx >> 9, f = idx & 511;
  float b1v = b1[0];
  const float* Wqh = Wq + (size_t)h * E_ * E_;
  float acc = 0.f;
  for (int e = 0; e < E_; ++e)
    acc += (b1v * Wl[e] + bl[e]) * Wqh[(size_t)e * E_ + f];
  qp[idx] = acc;
}

__global__ __launch_bounds__(256) void dec_wt(const float* __restrict__ Wk,
                                              const float* __restrict__ qp,
                                              float* __restrict__ wt) {
  int idx = blockIdx.x * 256 + threadIdx.x;    // H*E
  int h = idx >> 9, e = idx & 511;
  const float* row = Wk + ((size_t)h * E_ + e) * E_;
  const float* q = qp + h * E_;
  float acc = 0.f;
  for (int f = 0; f < E_; ++f) acc += row[f] * q[f];
  wt[idx] = acc;
}

__global__ __launch_bounds__(256) void dec_scores(const float* __restrict__ X,
                                                  const float* __restrict__ wt,
                                                  float* __restrict__ S) {
  int idx = blockIdx.x * 256 + threadIdx.x;    // B*H*N
  int n = idx & (N_ - 1);
  int bh = idx >> 9;
  int h = bh & (H_ - 1);
  int b = bh >> 3;
  const float* xr = X + ((size_t)b * N_ + n) * E_;
  const float* w = wt + h * E_;
  float acc = 0.f;
  for (int e = 0; e < E_; ++e) acc += xr[e] * w[e];
  S[idx] = acc;
}

__global__ __launch_bounds__(256) void dec_xbar(const float* __restrict__ X,
                                                const float* __restrict__ A,
                                                float* __restrict__ xbar) {
  int e = blockIdx.x * 256 + threadIdx.x;
  int bh = blockIdx.y;
  int b = bh >> 3;
  const float* a = A + (size_t)bh * N_;
  float acc = 0.f;
  for (int n = 0; n < N_; ++n) acc += a[n] * X[((size_t)b * N_ + n) * E_ + e];
  xbar[(size_t)bh * E_ + e] = acc;
}

__global__ __launch_bounds__(256) void dec_vproj(const float* __restrict__ xbar,
                                                 const float* __restrict__ Wv,
                                                 float* __restrict__ vo) {
  int f = blockIdx.x * 256 + threadIdx.x;
  int bh = blockIdx.y;
  int b = bh >> 3, h = bh & 7;
  const float* xr = xbar + (size_t)bh * E_;
  const float* Wvh = Wv + (size_t)h * E_ * E_;
  float acc = 0.f;
  for (int e = 0; e < E_; ++e) acc += xr[e] * Wvh[(size_t)e * E_ + f];
  vo[(size_t)b * HE_ + h * E_ + f] = acc;
}

__global__ __launch_bounds__(256) void dec_wo(const float* __restrict__ vo,
                                              const float* __restrict__ Wo,
                                              const float* __restrict__ Wl,
                                              const float* __restrict__ bl,
                                              const float* __restrict__ b1,
                                              float* __restrict__ out) {
  int e = blockIdx.x * 256 + threadIdx.x;
  int b = blockIdx.y;
  const float* v = vo + (size_t)b * HE_;
  float acc = 0.f;
  for (int j = 0; j < HE_; ++j) acc += v[j] * Wo[(size_t)j * E_ + e];
  out[(size_t)b * E_ + e] = acc + (b1[0] * Wl[e] + bl[e]);   // + residual Q
}

__global__ __launch_bounds__(256) void dec_ffn1(const float* __restrict__ x2,
                                                const float* __restrict__ W1,
                                                const float* __restrict__ bf1,
                                                float* __restrict__ hid) {
  int f = blockIdx.x * 256 + threadIdx.x;
  int b = blockIdx.y;
  const float* xr = x2 + (size_t)b * E_;
  float acc = bf1[f];
  for (int e = 0; e < E_; ++e) acc += xr[e] * W1[(size_t)e * F_ + f];
  hid[(size_t)b * F_ + f] = fmaxf(acc, 0.f);
}

__global__ __launch_bounds__(256) void dec_ffn2(const float* __restrict__ hid,
                                                const float* __restrict__ W2,
                                                const float* __restrict__ bf2,
                                                const float* __restrict__ x2,
                                                float* __restrict__ out) {
  int e = blockIdx.x * 256 + threadIdx.x;
  int b = blockIdx.y;
  const float* hr = hid + (size_t)b * F_;
  float acc = bf2[e] + x2[(size_t)b * E_ + e];
  for (int f = 0; f < F_; ++f) acc += hr[f] * W2[(size_t)f * E_ + e];
  out[(size_t)b * E_ + e] = acc;
}

__global__ __launch_bounds__(256) void bcast_rows(const float* __restrict__ rowv,
                                                  float* __restrict__ out) {
  long long idx = (long long)blockIdx.x * 256 + threadIdx.x;  // B*N*E
  int e = (int)(idx & (E_ - 1));
  int b = (int)(idx >> 18);                                   // / (N_*E_)
  out[idx] = rowv[(size_t)b * E_ + e];
}

__global__ __launch_bounds__(256) void final_proj(const float* __restrict__ X,
                                                  const float* __restrict__ W,
                                                  const float* __restrict__ b0,
                                                  float* __restrict__ O) {
  int r = blockIdx.x * 256 + threadIdx.x;   // ROWS_
  const float* xr = X + (size_t)r * E_;
  float acc = b0[0];
  for (int e = 0; e < E_; ++e) acc += xr[e] * W[e];
  O[r] = 1.0f / (1.0f + expf(-acc));
}

// ------------------------------- host side ---------------------------------
static inline void launch_gemm_nn(dim3 grid, hipStream_t stream,
    const float* A, long long sAz, int lda,
    const float* B, long long sBz, int ldb,
    float* C, long long sCz, int ldc,
    const float* bias, const float* resid, long long sRz, int ldr,
    int K, int relu) {
#if HAVE_ASYNC_LDS
  gemm_nn_async<<<grid, dim3(256), 0, stream>>>(
      A, sAz, lda, B, sBz, ldb, C, sCz, ldc, bias, resid, sRz, ldr, K, relu);
#else
  gemm_f32_wmma<0><<<grid, dim3(256), 0, stream>>>(
      A, sAz, lda, B, sBz, ldb, C, sCz, ldc, bias, resid, sRz, ldr, K, relu);
#endif
}

extern "C" void kernel_launch(void* const* d_in, const int* in_sizes, int n_in,
                              void* d_out, int out_size, void* d_ws, size_t ws_size,
                              hipStream_t stream) {
  (void)in_sizes; (void)n_in; (void)out_size; (void)ws_size;

  // inputs per setup_inputs() flattening order
  const float* x_in = (const float*)d_in[0];
  // d_in[1] = y : provably unused (size-1 LayerNorm kills the trg stream)
  const float* eWq  = (const float*)d_in[2];
  const float* eWk  = (const float*)d_in[3];
  const float* eWv  = (const float*)d_in[4];
  const float* eWo  = (const float*)d_in[5];
  const float* eg1  = (const float*)d_in[6];
  const float* eb1  = (const float*)d_in[7];
  const float* eg2  = (const float*)d_in[8];
  const float* eb2  = (const float*)d_in[9];
  const float* eW1  = (const float*)d_in[10];
  const float* ebf1 = (const float*)d_in[11];
  const float* eW2  = (const float*)d_in[12];
  const float* ebf2 = (const float*)d_in[13];
  // d_in[14..18]: dec mWq/mWk/mWv/mWo/g1 : provably unused
  const float* db1  = (const float*)d_in[19];
  const float* dWl  = (const float*)d_in[20];
  const float* dbl  = (const float*)d_in[21];
  const float* dWq  = (const float*)d_in[22];
  const float* dWk  = (const float*)d_in[23];
  const float* dWv  = (const float*)d_in[24];
  const float* dWo  = (const float*)d_in[25];
  const float* dg2  = (const float*)d_in[26];
  const float* db2  = (const float*)d_in[27];
  const float* dg3  = (const float*)d_in[28];
  const float* db3  = (const float*)d_in[29];
  const float* dW1  = (const float*)d_in[30];
  const float* dbf1 = (const float*)d_in[31];
  const float* dW2  = (const float*)d_in[32];
  const float* dbf2 = (const float*)d_in[33];
  const float* oW   = (const float*)d_in[34];
  const float* ob   = (const float*)d_in[35];

  // workspace carve (floats); total ~118 MB
  float* w = (float*)d_ws;
  size_t off = 0;
  auto take = [&](size_t n) { float* p = w + off; off += n; return p; };
  float* xa   = take((size_t)ROWS_ * E_);
  float* xb   = take((size_t)ROWS_ * E_);
  float* qh   = take((size_t)ROWS_ * E_);
  float* kh   = take((size_t)ROWS_ * E_);
  float* vh   = take((size_t)ROWS_ * E_);
  float* sc   = take((size_t)B_ * N_ * N_);
  float* cat  = take((size_t)ROWS_ * HE_);
  float* hid  = cat;                       // alias: cat is dead when FFN runs
  float* qp    = take(HE_);
  float* wt    = take(HE_);
  float* dsc   = take((size_t)B_ * H_ * N_);
  float* xbar  = take((size_t)B_ * H_ * E_);
  float* vo    = take((size_t)B_ * HE_);
  float* x2pre = take((size_t)B_ * E_);
  float* x2    = take((size_t)B_ * E_);
  float* dhid  = take((size_t)B_ * F_);
  float* dlpre = take((size_t)B_ * E_);
  float* dl    = take((size_t)B_ * E_);

  const dim3 blk(256);
  const float inv_e = 1.0f / (float)E_;    // reference divides scores by E (not sqrt)
  auto ggrid = [](int M, int Nc, int z) { return dim3(Nc / 64, M / 128, z); };

  // x = x + pos_enc
  posenc_add<<<(ROWS_ * E_) / 256, blk, 0, stream>>>(x_in, xa);
  float* cur = xa;

  // ------------------------------ encoder ------------------------------
  for (int l = 0; l < 2; ++l) {
    const float* Wq_l  = eWq + (size_t)l * H_ * E_ * E_;
    const float* Wk_l  = eWk + (size_t)l * H_ * E_ * E_;
    const float* Wv_l  = eWv + (size_t)l * H_ * E_ * E_;
    const float* Wo_l  = eWo + (size_t)l * HE_ * E_;
    const float* W1_l  = eW1 + (size_t)l * E_ * F_;
    const float* W2_l  = eW2 + (size_t)l * F_ * E_;

    for (int h = 0; h < H_; ++h) {
      const size_t ho = (size_t)h * E_ * E_;
      // per-head full-width projections: [ROWS_,E] @ [E,E]
      launch_gemm_nn(ggrid(ROWS_, E_, 1), stream,
          cur, 0, E_, Wq_l + ho, 0, E_, qh, 0, E_, nullptr, nullptr, 0, 0, E_, 0);
      launch_gemm_nn(ggrid(ROWS_, E_, 1), stream,
          cur, 0, E_, Wk_l + ho, 0, E_, kh, 0, E_, nullptr, nullptr, 0, 0, E_, 0);
      launch_gemm_nn(ggrid(ROWS_, E_, 1), stream,
          cur, 0, E_, Wv_l + ho, 0, E_, vh, 0, E_, nullptr, nullptr, 0, 0, E_, 0);
      // scores = Q @ K^T  (batched over b; NT needs LDS transpose -> VGPR path)
      gemm_f32_wmma<1><<<ggrid(N_, N_, B_), blk, 0, stream>>>(
          qh, (long long)N_ * E_, E_, kh, (long long)N_ * E_, E_,
          sc, (long long)N_ * N_, N_, nullptr, nullptr, 0, 0, E_, 0);
      softmax_rows<<<B_ * N_, blk, 0, stream>>>(sc, N_, inv_e);
      // out_h = A @ V  -> concat column slice h*E
      launch_gemm_nn(ggrid(N_, E_, B_), stream,
          sc, (long long)N_ * N_, N_, vh, (long long)N_ * E_, E_,
          cat + (size_t)h * E_, (long long)N_ * HE_, HE_,
          nullptr, nullptr, 0, 0, N_, 0);
    }
    // Wo projection + residual, then LN(g1,b1)
    launch_gemm_nn(ggrid(ROWS_, E_, 1), stream,
        cat, 0, HE_, Wo_l, 0, E_, xb, 0, E_, nullptr, cur, 0, E_, HE_, 0);
    layernorm512<<<ROWS_, blk, 0, stream>>>(xb, xb, eg1 + l * E_, eb1 + l * E_);
    // FFN
    launch_gemm_nn(ggrid(ROWS_, F_, 1), stream,
        xb, 0, E_, W1_l, 0, F_, hid, 0, F_, ebf1 + l * F_, nullptr, 0, 0, E_, 1);
    launch_gemm_nn(ggrid(ROWS_, E_, 1), stream,
        hid, 0, F_, W2_l, 0, E_, cur, 0, E_, ebf2 + l * E_, xb, 0, E_, F_, 0);
    layernorm512<<<ROWS_, blk, 0, stream>>>(cur, cur, eg2 + l * E_, eb2 + l * E_);
  }

  // ------------------------------ decoder ------------------------------
  float* xsrc = cur;      // K = V = previous stream
  float* xdst = (cur == xa) ? xb : xa;
  for (int l = 0; l < 2; ++l) {
    const float* Wq_l = dWq + (size_t)l * H_ * E_ * E_;
    const float* Wk_l = dWk + (size_t)l * H_ * E_ * E_;
    const float* Wv_l = dWv + (size_t)l * H_ * E_ * E_;
    const float* Wo_l = dWo + (size_t)l * HE_ * E_;
    const float* Wl_l = dWl + (size_t)l * E_;
    const float* bl_l = dbl + (size_t)l * E_;
    const float* b1_l = db1 + l;

    dec_qp<<<HE_ / 256, blk, 0, stream>>>(Wq_l, Wl_l, bl_l, b1_l, qp);
    dec_wt<<<HE_ / 256, blk, 0, stream>>>(Wk_l, qp, wt);
    dec_scores<<<(B_ * H_ * N_) / 256, blk, 0, stream>>>(xsrc, wt, dsc);
    softmax_rows<<<B_ * H_, blk, 0, stream>>>(dsc, N_, inv_e);
    dec_xbar<<<dim3(E_ / 256, B_ * H_), blk, 0, stream>>>(xsrc, dsc, xbar);
    dec_vproj<<<dim3(E_ / 256, B_ * H_), blk, 0, stream>>>(xbar, Wv_l, vo);
    dec_wo<<<dim3(E_ / 256, B_), blk, 0, stream>>>(vo, Wo_l, Wl_l, bl_l, b1_l, x2pre);
    layernorm512<<<B_, blk, 0, stream>>>(x2pre, x2, dg2 + l * E_, db2 + l * E_);
    dec_ffn1<<<dim3(F_ / 256, B_), blk, 0, stream>>>(x2, dW1 + (size_t)l * E_ * F_,
                                                     dbf1 + l * F_, dhid);
    dec_ffn2<<<dim3(E_ / 256, B_), blk, 0, stream>>>(dhid, dW2 + (size_t)l * F_ * E_,
                                                     dbf2 + l * E_, x2, dlpre);
    layernorm512<<<B_, blk, 0, stream>>>(dlpre, dl, dg3 + l * E_, db3 + l * E_);
    bcast_rows<<<(ROWS_ * E_) / 256, blk, 0, stream>>>(dl, xdst);
    float* t = xsrc; xsrc = xdst; xdst = t;
  }

  // final projection + sigmoid
  final_proj<<<ROWS_ / 256, blk, 0, stream>>>(xsrc, oW, ob, (float*)d_out);
}